// MultiHashEncoding_22076131901758
// MI455X (gfx1250) — compile-verified
//
#include <hip/hip_runtime.h>
#include <cstdint>

#define TPB    256
#define CHUNKS 4
#define NPTS   1048576
#define NBLK   (NPTS / (TPB * CHUNKS))   // 1024 blocks

// ---------------- CDNA5 async-to-LDS plumbing (compile-guarded) -------------
#if defined(__HIP_DEVICE_COMPILE__)
  #if __has_builtin(__builtin_amdgcn_global_load_async_to_lds_b32)
    #define ASYNC_MODE 1   // clang builtin path
  #else
    #define ASYNC_MODE 2   // inline-asm path (ISA mnemonic from CDNA5 ISA)
  #endif
#else
  #define ASYNC_MODE 0     // host pass: parse-only fallback
#endif

#if ASYNC_MODE
  #if __has_builtin(__builtin_amdgcn_s_wait_asynccnt)
    #define WAIT_ASYNC(n) __builtin_amdgcn_s_wait_asynccnt(n)
  #else
    #define WAIT_ASYNC(n) asm volatile("s_wait_asynccnt " #n ::: "memory")
  #endif
#else
  #define WAIT_ASYNC(n)
#endif

#if ASYNC_MODE == 1
typedef __attribute__((address_space(1))) int gmem_i;   // printed as '__device__ int' by clang
typedef __attribute__((address_space(3))) int lds_i;
#endif

__device__ __forceinline__ void async_ld_f32(const float* g, float* l) {
#if ASYNC_MODE == 1
  __builtin_amdgcn_global_load_async_to_lds_b32(
      (gmem_i*)(uintptr_t)g,
      (lds_i*)(uintptr_t)(uint32_t)(uintptr_t)l,
      /*offset=*/0, /*cpol=*/0);
#elif ASYNC_MODE == 2
  uint32_t lo = (uint32_t)(uintptr_t)l;             // LDS byte offset
  asm volatile("global_load_async_to_lds_b32 %0, %1, off"
               :: "v"(lo), "v"((uint64_t)(uintptr_t)g) : "memory");
#else
  (void)g; (void)l;
#endif
}

// ---------------- per-level trilinear gather/blend --------------------------
#define ACC4(A, V)                                          \
  A.x = fmaf(w, V.x, A.x); A.y = fmaf(w, V.y, A.y);         \
  A.z = fmaf(w, V.z, A.z); A.w = fmaf(w, V.w, A.w);

template <int G0, int G1, int G2>
__device__ __forceinline__ void encode_level(const float* __restrict__ emb,
                                             float px, float py, float pz,
                                             float* __restrict__ dst) {
  const float cx = px * (float)(G0 - 1);
  const float cy = py * (float)(G1 - 1);
  const float cz = pz * (float)(G2 - 1);
  const float fx = floorf(cx), fy = floorf(cy), fz = floorf(cz);

  int ix[2], iy[2], iz[2];
  ix[0] = min(max((int)fx, 0), G0 - 1); ix[1] = min((int)fx + 1, G0 - 1);
  iy[0] = min(max((int)fy, 0), G1 - 1); iy[1] = min((int)fy + 1, G1 - 1);
  iz[0] = min(max((int)fz, 0), G2 - 1); iz[1] = min((int)fz + 1, G2 - 1);

  float wx[2], wy[2], wz[2];
  wx[0] = 1.0f - fabsf((float)ix[0] - cx); wx[1] = 1.0f - fabsf((float)ix[1] - cx);
  wy[0] = 1.0f - fabsf((float)iy[0] - cy); wy[1] = 1.0f - fabsf((float)iy[1] - cy);
  wz[0] = 1.0f - fabsf((float)iz[0] - cz); wz[1] = 1.0f - fabsf((float)iz[1] - cz);

  float4 a0 = {0,0,0,0}, a1 = {0,0,0,0}, a2 = {0,0,0,0}, a3 = {0,0,0,0};

#pragma unroll
  for (int a = 0; a < 2; ++a) {
#pragma unroll
    for (int b = 0; b < 2; ++b) {
#pragma unroll
      for (int c = 0; c < 2; ++c) {
        const float w  = wx[a] * wy[b] * wz[c];
        const int idx  = ((ix[a] * G1 + iy[b]) * G2 + iz[c]) * 16;
        const float4* e = (const float4*)(emb + idx);   // 64B row, aligned -> 4x b128
        float4 v0 = e[0], v1 = e[1], v2 = e[2], v3 = e[3];
        ACC4(a0, v0) ACC4(a1, v1) ACC4(a2, v2) ACC4(a3, v3)
      }
    }
  }
  dst[0]  = a0.x; dst[1]  = a0.y; dst[2]  = a0.z; dst[3]  = a0.w;
  dst[4]  = a1.x; dst[5]  = a1.y; dst[6]  = a1.z; dst[7]  = a1.w;
  dst[8]  = a2.x; dst[9]  = a2.y; dst[10] = a2.z; dst[11] = a2.w;
  dst[12] = a3.x; dst[13] = a3.y; dst[14] = a3.z; dst[15] = a3.w;
}

// ---------------- main kernel ----------------------------------------------
__global__ void __launch_bounds__(TPB)
mhe_kernel(const float* __restrict__ in, const float* __restrict__ emb0,
           const float* __restrict__ emb1, float* __restrict__ out) {
  __shared__ float s_in[2][TPB * 3];   // double-buffered coords (async target)
  __shared__ float s_out[TPB * 36];    // padded rows -> coalesced drain

  const int tid = threadIdx.x;
  const int blk = blockIdx.x;

#if ASYNC_MODE
  { // kick off chunk 0 coords
    const float* g = in + (size_t)blk * (CHUNKS * TPB * 3) + tid;
    async_ld_f32(g,            &s_in[0][tid]);
    async_ld_f32(g + TPB,      &s_in[0][tid + TPB]);
    async_ld_f32(g + 2 * TPB,  &s_in[0][tid + 2 * TPB]);
  }
#endif

  for (int c = 0; c < CHUNKS; ++c) {
    const int P   = (blk * CHUNKS + c) * TPB;   // first point of this chunk
    const int buf = c & 1;

#if ASYNC_MODE
    if (c + 1 < CHUNKS) {  // prefetch next chunk's coords while gathers run
      const float* g = in + (size_t)(P + TPB) * 3 + tid;
      async_ld_f32(g,           &s_in[buf ^ 1][tid]);
      async_ld_f32(g + TPB,     &s_in[buf ^ 1][tid + TPB]);
      async_ld_f32(g + 2 * TPB, &s_in[buf ^ 1][tid + 2 * TPB]);
      WAIT_ASYNC(3);   // retire the older triple (in-order completion)
    } else {
      WAIT_ASYNC(0);
    }
    __syncthreads();   // coords visible block-wide; also guards s_out reuse
    const float px = s_in[buf][tid * 3 + 0];
    const float py = s_in[buf][tid * 3 + 1];
    const float pz = s_in[buf][tid * 3 + 2];
#else
    __syncthreads();   // guard s_out reuse
    if (c + 1 < CHUNKS)
      __builtin_prefetch(in + (size_t)(P + TPB + tid) * 3, 0, 0); // global_prefetch_b8
    const float* cp = in + (size_t)(P + tid) * 3;
    const float px = cp[0], py = cp[1], pz = cp[2];
#endif

    float* row = &s_out[tid * 36];
    row[0] = px; row[1] = py; row[2] = pz;
    encode_level<5, 91, 161>(emb0, px, py, pz, row + 3);   // L2-resident 4.7MB table
    encode_level<3, 46, 81>(emb1, px, py, pz, row + 19);   // L2-resident 0.7MB table
    __syncthreads();

    // Coalesced drain: TPB*35 floats = 2240 float4s, base 8960*k floats (16B aligned)
    float4* ob = (float4*)(out + (size_t)P * 35);
    for (int i4 = tid; i4 < TPB * 35 / 4; i4 += TPB) {
      const int f = i4 * 4;
      float4 v;
      v.x = s_out[((f + 0) / 35) * 36 + (f + 0) % 35];
      v.y = s_out[((f + 1) / 35) * 36 + (f + 1) % 35];
      v.z = s_out[((f + 2) / 35) * 36 + (f + 2) % 35];
      v.w = s_out[((f + 3) / 35) * 36 + (f + 3) % 35];
      ob[i4] = v;
    }
  }
}

// ---------------- launcher --------------------------------------------------
extern "C" void kernel_launch(void* const* d_in, const int* in_sizes, int n_in,
                              void* d_out, int out_size, void* d_ws, size_t ws_size,
                              hipStream_t stream) {
  (void)n_in; (void)out_size; (void)d_ws; (void)ws_size;
  const float* in   = (const float*)d_in[0];
  const float* emb0 = (const float*)d_in[1];
  const float* emb1 = (const float*)d_in[2];
  float* out = (float*)d_out;

  const int npts   = in_sizes[0] / 3;           // 1048576
  const int blocks = npts / (TPB * CHUNKS);     // 1024
  hipLaunchKernelGGL(mhe_kernel, dim3(blocks), dim3(TPB), 0, stream,
                     in, emb0, emb1, out);
}